// DyadicSignatures_31250182045837
// MI455X (gfx1250) — compile-verified
//
#include <hip/hip_runtime.h>

// ---------------------------------------------------------------------------
// DyadicSignatures for MI455X (gfx1250)
//   - one wave (32 lanes) per path; one lane per 32-frame leaf segment
//   - leaf scans + Chen-relation tree combine via wave32 __shfl_xor
//   - path data gathered by the Tensor Data Mover (tensor_load_to_lds) with
//     LDS padding (16B per 512B) to kill bank conflicts; s_wait_tensorcnt
// ---------------------------------------------------------------------------

typedef __attribute__((ext_vector_type(4))) unsigned int v4u_t;
typedef __attribute__((ext_vector_type(8))) int          v8i_t;
typedef __attribute__((ext_vector_type(4))) int          v4i_t;

#define PATHS_PER_BLOCK 2
#define FRAMES          1024
#define SEG_FRAMES      32
// padded LDS layout: frame t (float4 units) lives at  t + (t >> 5)
#define PATH_F4         (FRAMES + FRAMES / SEG_FRAMES)   // 1056 float4 = 16896 B

__device__ __forceinline__ int faddr(int t) { return t + (t >> 5); }

// S <- S (x) exp(dx): exact reference scan step (update s3, then s2, then s1)
// layout: s[0..3]=s1, s[4..19]=s2 (row-major), s[20..83]=s3 (row-major)
__device__ __forceinline__ void sig_step(float* __restrict__ s,
                                         const float* __restrict__ dx) {
  float dx2[16];
#pragma unroll
  for (int a = 0; a < 4; ++a)
#pragma unroll
    for (int b = 0; b < 4; ++b) dx2[a * 4 + b] = dx[a] * dx[b];
  float t2[16];
#pragma unroll
  for (int i = 0; i < 16; ++i) t2[i] = fmaf(dx2[i], 0.16666666666666666f, s[4 + i]);
  float h1[4];
#pragma unroll
  for (int a = 0; a < 4; ++a) h1[a] = 0.5f * s[a];
#pragma unroll
  for (int a = 0; a < 4; ++a)
#pragma unroll
    for (int b = 0; b < 4; ++b)
#pragma unroll
      for (int c = 0; c < 4; ++c) {
        const int idx = 20 + (a * 4 + b) * 4 + c;
        s[idx] = fmaf(t2[a * 4 + b], dx[c], fmaf(h1[a], dx2[b * 4 + c], s[idx]));
      }
#pragma unroll
  for (int a = 0; a < 4; ++a)
#pragma unroll
    for (int b = 0; b < 4; ++b) {
      const int idx = 4 + a * 4 + b;
      s[idx] = fmaf(s[a], dx[b], fmaf(0.5f, dx2[a * 4 + b], s[idx]));
    }
#pragma unroll
  for (int a = 0; a < 4; ++a) s[a] += dx[a];
}

// Chen product in place: s <- s (x) o   (s = left signature, o = right)
__device__ __forceinline__ void chen_inplace(float* __restrict__ s,
                                             const float* __restrict__ o) {
#pragma unroll
  for (int a = 0; a < 4; ++a)
#pragma unroll
    for (int b = 0; b < 4; ++b)
#pragma unroll
      for (int c = 0; c < 4; ++c) {
        const int idx = 20 + (a * 4 + b) * 4 + c;
        float t = s[idx] + o[idx];
        t = fmaf(s[a], o[4 + b * 4 + c], t);
        t = fmaf(s[4 + a * 4 + b], o[c], t);
        s[idx] = t;
      }
#pragma unroll
  for (int a = 0; a < 4; ++a)
#pragma unroll
    for (int b = 0; b < 4; ++b) {
      const int idx = 4 + a * 4 + b;
      s[idx] = fmaf(s[a], o[b], s[idx] + o[idx]);
    }
#pragma unroll
  for (int a = 0; a < 4; ++a) s[a] += o[a];
}

__global__ __launch_bounds__(64) void DyadicSignatures_gfx1250_kernel(
    const float* __restrict__ x, float* __restrict__ out) {
  __shared__ float lds[PATHS_PER_BLOCK * PATH_F4 * 4];  // 33792 B

  const int wave = threadIdx.x >> 5;
  const int lane = threadIdx.x & 31;
  const int path = blockIdx.x * PATHS_PER_BLOCK + wave;  // 0..4095
  const int b = path >> 6;
  const int e = path & 63;

  float* F = &lds[wave * PATH_F4 * 4];

  // ---- stage x[b, :, e, :] (1024 rows x 16B, 1KB apart) into LDS ----------
#if __has_builtin(__builtin_amdgcn_tensor_load_to_lds)
  {
    const unsigned int lds_byte =
        (unsigned int)(unsigned long long)(const void*)F;
    const unsigned long long ga = (unsigned long long)(const void*)x +
                                  (unsigned long long)b * 1048576ull +  // b*T*E*D*4
                                  (unsigned long long)e * 16ull;        // e*D*4
    // D# group 0: count=1 | lds_addr | global_addr | type=2
    v4u_t g0;
    g0[0] = 1u;
    g0[1] = lds_byte;
    g0[2] = (unsigned int)(ga & 0xffffffffull);
    g0[3] = (unsigned int)((ga >> 32) & 0x01ffffffull) | (2u << 30);
    // D# group 1: data_size=4B, pad 4 dwords every 128 dwords (16B per 512B),
    //             tensor/tile = [dim0=4, dim1=1024], dim0_stride = 256 elems
    v8i_t g1;
    g1[0] = (int)((2u << 16) | (1u << 20) | (6u << 22) | (3u << 25));
    g1[1] = (int)(4u << 16);       // tensor_dim0 = 4        (bits 63:48)
    g1[2] = (int)(1024u << 16);    // tensor_dim1 = 1024     (bits 95:80)
    g1[3] = (int)(4u << 16);       // tile_dim0   = 4        (bits 127:112)
    g1[4] = 1024;                  // tile_dim1   = 1024     (bits 143:128)
    g1[5] = 256;                   // tensor_dim0_stride=256 (bits 191:160)
    g1[6] = 0;
    g1[7] = 0;
    v4i_t g2 = {0, 0, 0, 0};
    v4i_t g3 = {0, 0, 0, 0};
#if defined(__clang_major__) && (__clang_major__ >= 23)
    v8i_t gz = {0, 0, 0, 0, 0, 0, 0, 0};
    __builtin_amdgcn_tensor_load_to_lds(g0, g1, g2, g3, gz, 0);
#else
    __builtin_amdgcn_tensor_load_to_lds(g0, g1, g2, g3, 0);
#endif
#if __has_builtin(__builtin_amdgcn_s_wait_tensorcnt)
    __builtin_amdgcn_s_wait_tensorcnt(0);
#endif
  }
#else
  {
    float4* dst = reinterpret_cast<float4*>(F);
    for (int t = lane; t < FRAMES; t += 32) {
      const float4 v = *reinterpret_cast<const float4*>(
          x + (((size_t)b * FRAMES + t) * 64 + e) * 4);
      dst[faddr(t)] = v;
    }
  }
#endif
  __syncthreads();
  asm volatile("" ::: "memory");

  const float4* F4 = reinterpret_cast<const float4*>(F);

  // ---- leaf scan: lane L owns frames [32L, 32L+32), 31 increments ---------
  float S[84];
#pragma unroll
  for (int i = 0; i < 84; ++i) S[i] = 0.0f;

  float4 prev = F4[faddr(lane * SEG_FRAMES)];
  for (int k = 1; k < SEG_FRAMES; ++k) {
    const float4 cur = F4[faddr(lane * SEG_FRAMES + k)];
    const float dx[4] = {cur.x - prev.x, cur.y - prev.y, cur.z - prev.z,
                         cur.w - prev.w};
    sig_step(S, dx);
    prev = cur;
  }

  // ---- Chen tree: 5 rounds; emit dyadic pieces after rounds 2..5 ----------
  float O[84];
  for (int r = 1; r <= 5; ++r) {
    const int m = 1 << (r - 1);
#pragma unroll
    for (int i = 0; i < 84; ++i) O[i] = __shfl_xor(S[i], m, 32);
    if ((lane & m) != 0) {  // I hold the right block: swap so S = left
#pragma unroll
      for (int i = 0; i < 84; ++i) {
        const float t = S[i];
        S[i] = O[i];
        O[i] = t;
      }
    }
    // boundary increment between the two blocks
    const int mseg = ((lane >> r) << r) + m;
    const int mf = mseg * SEG_FRAMES;
    const float4 fb = F4[faddr(mf)];
    const float4 fa = F4[faddr(mf - 1)];
    const float d[4] = {fb.x - fa.x, fb.y - fa.y, fb.z - fa.z, fb.w - fa.w};
    sig_step(S, d);      // S <- S (x) exp(d)
    chen_inplace(S, O);  // S <- S (x) O   (signature of merged block)

    if (r >= 2 && (lane & ((1 << r) - 1)) == 0) {
      // pieces: r=2 -> 7..14, r=3 -> 3..6, r=4 -> 1..2, r=5 -> 0
      const int piece = ((1 << (5 - r)) - 1) + (lane >> r);
      float* o = out + ((size_t)path * 15 + (size_t)piece) * 84;
#pragma unroll
      for (int i = 0; i < 84; ++i) o[i] = S[i];
    }
  }
}

extern "C" void kernel_launch(void* const* d_in, const int* in_sizes, int n_in,
                              void* d_out, int out_size, void* d_ws,
                              size_t ws_size, hipStream_t stream) {
  (void)in_sizes; (void)n_in; (void)out_size; (void)d_ws; (void)ws_size;
  const float* x = (const float*)d_in[0];
  float* out = (float*)d_out;
  // 4096 paths / 2 paths-per-block = 2048 blocks of 64 threads (2 waves)
  hipLaunchKernelGGL(DyadicSignatures_gfx1250_kernel, dim3(2048), dim3(64), 0,
                     stream, x, out);
}